// MLA_86990267613376
// MI455X (gfx1250) — compile-verified
//
#include <hip/hip_runtime.h>
#include <math.h>

typedef __attribute__((ext_vector_type(2))) float v2f;
typedef __attribute__((ext_vector_type(8))) float v8f;

#define ROPE_DIM 64
#define KV_LORA 512
#define KV_W    576      // KV_LORA + ROPE_DIM
#define HEADS   128
#define QL      1536
#define DIM     7168
#define VDIM    128
#define SCALE_QK 0.07216878364870323f  // 1/sqrt(128+64)
#define LOG1E4_OVER_32 0.2878231366242557f  // ln(10000)/32

__device__ __forceinline__ v8f wmma4(v2f a, v2f b, v8f c) {
  // D = A(16x4 f32) x B(4x16 f32) + C(16x16 f32)
  return __builtin_amdgcn_wmma_f32_16x16x4_f32(
      /*neg_a=*/false, a, /*neg_b=*/false, b,
      /*c_mod=*/(short)0, c, /*reuse_a=*/false, /*reuse_b=*/false);
}

// ---------------------------------------------------------------------------
// Generic GEMM: C[M,N] = A[M,K] (row-major) * B[K,N] (row-major)
// One wave owns MT consecutive 16-row M-tiles x one 16-col N-tile, so each
// B fragment is loaded once and feeds MT WMMAs (halves weight HBM traffic
// for MT=2 on the M=32 GEMMs). Optional batching via blockIdx.y + strides.
// ---------------------------------------------------------------------------
template<int MT>
__global__ __launch_bounds__(128) void gemm_rrr_mt(
    const float* __restrict__ A, const float* __restrict__ B, float* __restrict__ C,
    int M, int N, int K, int lda, int ldb, int ldc,
    long sA, long sB, long sC)
{
  long batch = blockIdx.y;
  A += batch * sA; B += batch * sB; C += batch * sC;
  int wave = (blockIdx.x * blockDim.x + threadIdx.x) >> 5;
  int lane = threadIdx.x & 31;
  int nt = N >> 4;
  int mg = (M >> 4) / MT;
  if (wave >= mg * nt) return;                 // wave-uniform: EXEC stays all-1s
  int tm = (wave / nt) * (16 * MT);
  int tn = (wave % nt) << 4;
  int mrow = lane & 15;                        // M index for A-frag, N index for B/C-frag
  int half = lane >> 4;                        // K sub-half select

  const float* Ap = A + (long)(tm + mrow) * lda + half * 2;
  const float* Bp = B + (long)(half * 2) * ldb + tn + mrow;
  v8f acc[MT];
  #pragma unroll
  for (int t = 0; t < MT; ++t) acc[t] = (v8f){};
  for (int k = 0; k < K; k += 4) {
    v2f bv;
    bv.x = Bp[(long)k * ldb];  bv.y = Bp[(long)k * ldb + ldb];
    #pragma unroll
    for (int t = 0; t < MT; ++t) {
      v2f av;
      av.x = Ap[(long)(t * 16) * lda + k];
      av.y = Ap[(long)(t * 16) * lda + k + 1];
      acc[t] = wmma4(av, bv, acc[t]);
    }
  }
  #pragma unroll
  for (int t = 0; t < MT; ++t) {
    float* Cp = C + (long)(tm + t * 16 + half * 8) * ldc + tn + mrow;
    #pragma unroll
    for (int v = 0; v < 8; ++v) Cp[(long)v * ldc] = acc[t][v];
  }
}

// ---------------------------------------------------------------------------
// GEMM with B transposed: C[M,N] = A[M,K] (row-major) * Bt[N,K]^T
// Used for q_abs: einsum('bq,hdq->bhd')  (Bt = W_qk_absorb[h], N=d, K=q)
// B-fragment loads are contiguous (b64) here.
// ---------------------------------------------------------------------------
template<int MT>
__global__ __launch_bounds__(128) void gemm_rcr_mt(
    const float* __restrict__ A, const float* __restrict__ Bt, float* __restrict__ C,
    int M, int N, int K, int lda, int ldb, int ldc,
    long sA, long sB, long sC)
{
  long batch = blockIdx.y;
  A += batch * sA; Bt += batch * sB; C += batch * sC;
  int wave = (blockIdx.x * blockDim.x + threadIdx.x) >> 5;
  int lane = threadIdx.x & 31;
  int nt = N >> 4;
  int mg = (M >> 4) / MT;
  if (wave >= mg * nt) return;
  int tm = (wave / nt) * (16 * MT);
  int tn = (wave % nt) << 4;
  int mrow = lane & 15;
  int half = lane >> 4;

  const float* Ap = A  + (long)(tm + mrow) * lda + half * 2;
  const float* Bp = Bt + (long)(tn + mrow) * ldb + half * 2;
  v8f acc[MT];
  #pragma unroll
  for (int t = 0; t < MT; ++t) acc[t] = (v8f){};
  for (int k = 0; k < K; k += 4) {
    v2f bv;
    bv.x = Bp[k];  bv.y = Bp[k + 1];
    #pragma unroll
    for (int t = 0; t < MT; ++t) {
      v2f av;
      av.x = Ap[(long)(t * 16) * lda + k];
      av.y = Ap[(long)(t * 16) * lda + k + 1];
      acc[t] = wmma4(av, bv, acc[t]);
    }
  }
  #pragma unroll
  for (int t = 0; t < MT; ++t) {
    float* Cp = C + (long)(tm + t * 16 + half * 8) * ldc + tn + mrow;
    #pragma unroll
    for (int v = 0; v < 8; ++v) Cp[(long)v * ldc] = acc[t][v];
  }
}

// ---------------------------------------------------------------------------
// RoPE on q_pe (in place), fixed position S-1. Pair trick (i, i+32).
// ---------------------------------------------------------------------------
__global__ void rope_q(float* __restrict__ qpe, int B, int S)
{
  int idx = blockIdx.x * blockDim.x + threadIdx.x;
  int total = B * HEADS * (ROPE_DIM / 2);
  if (idx >= total) return;
  int i  = idx & 31;
  int bh = idx >> 5;
  float* p = qpe + (long)bh * ROPE_DIM;
  float x0 = p[i], x1 = p[i + 32];
  float f = __expf(-(float)i * LOG1E4_OVER_32);
  float ang = (float)(S - 1) * f;
  float c = __cosf(ang), sn = __sinf(ang);
  p[i]      = x0 * c - x1 * sn;
  p[i + 32] = x1 * c + x0 * sn;
}

// ---------------------------------------------------------------------------
// RoPE on k_pe -> kpe buffer (B,S,64). Row S-1 sourced from kv_new.
// ---------------------------------------------------------------------------
__global__ void rope_k(const float* __restrict__ kv_cache,
                       const float* __restrict__ kv_new,
                       float* __restrict__ kpe, int B, int S)
{
  long idx = (long)blockIdx.x * blockDim.x + threadIdx.x;
  long total = (long)B * S * (ROPE_DIM / 2);
  if (idx >= total) return;
  int  i  = (int)(idx & 31);
  long bs = idx >> 5;                  // b*S + s
  int  s  = (int)(bs % S);
  int  b  = (int)(bs / S);
  const float* src = (s == S - 1) ? (kv_new + (long)b * KV_W + KV_LORA)
                                  : (kv_cache + bs * KV_W + KV_LORA);
  float x0 = src[i], x1 = src[i + 32];
  float f = __expf(-(float)i * LOG1E4_OVER_32);
  float ang = (float)s * f;
  float c = __cosf(ang), sn = __sinf(ang);
  float* dst = kpe + bs * ROPE_DIM;
  dst[i]      = x0 * c - x1 * sn;
  dst[i + 32] = x1 * c + x0 * sn;
}

// ---------------------------------------------------------------------------
// scores[b,h,s] = (q_abs[b,h,:]·c_kv[b,s,:] + q_pe[b,h,:]·k_pe[b,s,:]) * SCALE
// One wave covers ALL 8 head-tiles (M=128) for one 16-wide s-tile: each
// kv fragment is loaded once and feeds 8 WMMAs -> kv_cache read once/pass.
// Row s=S-1 comes from kv_new (inputs must not be mutated).
// ---------------------------------------------------------------------------
__global__ __launch_bounds__(128) void mla_scores(
    const float* __restrict__ q_abs,    // (B,H,512)
    const float* __restrict__ q_pe,     // (B,H,64)
    const float* __restrict__ kv_cache, // (B,S,576)
    const float* __restrict__ kv_new,   // (B,576)
    const float* __restrict__ kpe,      // (B,S,64)
    float* __restrict__ scores,         // (B,H,S)
    int S)
{
  int b    = blockIdx.y;
  int wave = (blockIdx.x * blockDim.x + threadIdx.x) >> 5;
  int lane = threadIdx.x & 31;
  int nt   = S >> 4;
  if (wave >= nt) return;
  int tn = wave << 4;
  int mrow = lane & 15;
  int half = lane >> 4;

  int s = tn + mrow;                          // this lane's kv row (N index)
  const float* kvrow = (s == S - 1) ? (kv_new + (long)b * KV_W)
                                    : (kv_cache + ((long)b * S + s) * KV_W);
  const float* Ap = q_abs + ((long)b * HEADS + mrow) * KV_LORA + half * 2;
  v8f acc[8];
  #pragma unroll
  for (int t = 0; t < 8; ++t) acc[t] = (v8f){};

  for (int k = 0; k < KV_LORA; k += 4) {
    v2f bv;
    bv.x = kvrow[k + half * 2];  bv.y = kvrow[k + half * 2 + 1];
    #pragma unroll
    for (int t = 0; t < 8; ++t) {
      v2f av;
      av.x = Ap[(long)(t * 16) * KV_LORA + k];
      av.y = Ap[(long)(t * 16) * KV_LORA + k + 1];
      acc[t] = wmma4(av, bv, acc[t]);
    }
  }
  const float* Ap2   = q_pe + ((long)b * HEADS + mrow) * ROPE_DIM + half * 2;
  const float* kprow = kpe + ((long)b * S + s) * ROPE_DIM;
  for (int k = 0; k < ROPE_DIM; k += 4) {
    v2f bv;
    bv.x = kprow[k + half * 2];  bv.y = kprow[k + half * 2 + 1];
    #pragma unroll
    for (int t = 0; t < 8; ++t) {
      v2f av;
      av.x = Ap2[(long)(t * 16) * ROPE_DIM + k];
      av.y = Ap2[(long)(t * 16) * ROPE_DIM + k + 1];
      acc[t] = wmma4(av, bv, acc[t]);
    }
  }
  #pragma unroll
  for (int t = 0; t < 8; ++t) {
    float* Cp = scores + ((long)b * HEADS + t * 16 + half * 8) * S + tn + mrow;
    #pragma unroll
    for (int v = 0; v < 8; ++v) Cp[(long)v * S] = acc[t][v] * SCALE_QK;
  }
}

// ---------------------------------------------------------------------------
// Row softmax over S, in place. One block per (b,h) row.
// ---------------------------------------------------------------------------
__global__ __launch_bounds__(256) void softmax_rows(float* __restrict__ scores, int S)
{
  float* row = scores + (long)blockIdx.x * S;
  __shared__ float red[256];
  int t = threadIdx.x;
  float m = -INFINITY;
  for (int i = t; i < S; i += 256) m = fmaxf(m, row[i]);
  red[t] = m; __syncthreads();
  for (int off = 128; off; off >>= 1) {
    if (t < off) red[t] = fmaxf(red[t], red[t + off]);
    __syncthreads();
  }
  m = red[0]; __syncthreads();
  float sum = 0.f;
  for (int i = t; i < S; i += 256) {
    float e = __expf(row[i] - m);
    row[i] = e; sum += e;
  }
  red[t] = sum; __syncthreads();
  for (int off = 128; off; off >>= 1) {
    if (t < off) red[t] += red[t + off];
    __syncthreads();
  }
  float inv = 1.0f / red[0];
  for (int i = t; i < S; i += 256) row[i] *= inv;
}

// ---------------------------------------------------------------------------
// o_lat[b,h,d] = sum_s attn[b,h,s] * c_kv[b,s,d]  (M=H=128, N=512, K=S)
// One wave covers all 8 head-tiles for one 16-wide d-tile: each kv fragment
// feeds 8 WMMAs -> kv_cache read once for this pass too.
// ---------------------------------------------------------------------------
__global__ __launch_bounds__(128) void mla_olat(
    const float* __restrict__ attn,     // (B,H,S)
    const float* __restrict__ kv_cache, // (B,S,576)
    const float* __restrict__ kv_new,   // (B,576)
    float* __restrict__ o_lat,          // (B,H,512)
    int S)
{
  int b    = blockIdx.y;
  int wave = (blockIdx.x * blockDim.x + threadIdx.x) >> 5;
  int lane = threadIdx.x & 31;
  int nt   = KV_LORA >> 4;
  if (wave >= nt) return;
  int tn = wave << 4;
  int mrow = lane & 15;
  int half = lane >> 4;

  const float* Ap = attn + ((long)b * HEADS + mrow) * S + half * 2;
  int n = tn + mrow;                          // d index
  const float* cacheB = kv_cache + (long)b * S * KV_W;
  const float* newB   = kv_new + (long)b * KV_W;
  v8f acc[8];
  #pragma unroll
  for (int t = 0; t < 8; ++t) acc[t] = (v8f){};

  for (int k = 0; k < S; k += 4) {
    int s0 = k + half * 2;
    const float* r0 = (s0     == S - 1) ? newB : cacheB + (long)s0 * KV_W;
    const float* r1 = (s0 + 1 == S - 1) ? newB : cacheB + (long)(s0 + 1) * KV_W;
    v2f bv;
    bv.x = r0[n];  bv.y = r1[n];
    #pragma unroll
    for (int t = 0; t < 8; ++t) {
      v2f av;
      av.x = Ap[(long)(t * 16) * S + k];
      av.y = Ap[(long)(t * 16) * S + k + 1];
      acc[t] = wmma4(av, bv, acc[t]);
    }
  }
  #pragma unroll
  for (int t = 0; t < 8; ++t) {
    float* Cp = o_lat + ((long)b * HEADS + t * 16 + half * 8) * KV_LORA + tn + mrow;
    #pragma unroll
    for (int v = 0; v < 8; ++v) Cp[(long)v * KV_LORA] = acc[t][v];
  }
}

// ---------------------------------------------------------------------------
extern "C" void kernel_launch(void* const* d_in, const int* in_sizes, int n_in,
                              void* d_out, int out_size, void* d_ws, size_t ws_size,
                              hipStream_t stream)
{
  const float* x        = (const float*)d_in[0];
  const float* kv_cache = (const float*)d_in[1];
  const float* Wq_down  = (const float*)d_in[2];
  const float* Wkv_down = (const float*)d_in[3];
  const float* Wq_rope  = (const float*)d_in[4];
  const float* Wqk      = (const float*)d_in[5]; // (H,512,1536)
  const float* Wuv      = (const float*)d_in[6]; // (H,512,128)
  const float* Wo       = (const float*)d_in[7]; // (H*128,7168)

  int B = in_sizes[0] / DIM;            // 32
  int S = in_sizes[1] / (B * KV_W);     // 4096

  float* ws = (float*)d_ws;
  float* q_lora = ws;  ws += (size_t)B * QL;
  float* kv_new = ws;  ws += (size_t)B * KV_W;
  float* q_pe   = ws;  ws += (size_t)B * HEADS * ROPE_DIM;   // q_pe_raw then roped in place
  float* q_abs  = ws;  ws += (size_t)B * HEADS * KV_LORA;
  float* kpe    = ws;  ws += (size_t)B * S * ROPE_DIM;
  float* scores = ws;  ws += (size_t)B * HEADS * S;
  float* o_lat  = ws;  ws += (size_t)B * HEADS * KV_LORA;
  float* o_cat  = ws;  ws += (size_t)B * HEADS * VDIM;
  float* out    = (float*)d_out;

  #define WAVES(M, N, MT) ((((M) / 16) / (MT)) * ((N) / 16))
  #define BLKS(w) (((w) + 3) / 4)

  // 1) q_lora = x @ Wq_down   (32x7168 @ 7168x1536), MT=2 -> B read once
  gemm_rrr_mt<2><<<dim3(BLKS(WAVES(B, QL, 2)), 1), 128, 0, stream>>>(
      x, Wq_down, q_lora, B, QL, DIM, DIM, QL, QL, 0, 0, 0);

  // 2) kv_new = x @ Wkv_down  (32x7168 @ 7168x576)
  gemm_rrr_mt<2><<<dim3(BLKS(WAVES(B, KV_W, 2)), 1), 128, 0, stream>>>(
      x, Wkv_down, kv_new, B, KV_W, DIM, DIM, KV_W, KV_W, 0, 0, 0);

  // 3) q_pe_raw = q_lora @ Wq_rope (32x1536 @ 1536x8192)
  gemm_rrr_mt<2><<<dim3(BLKS(WAVES(B, HEADS * ROPE_DIM, 2)), 1), 128, 0, stream>>>(
      q_lora, Wq_rope, q_pe, B, HEADS * ROPE_DIM, QL,
      QL, HEADS * ROPE_DIM, HEADS * ROPE_DIM, 0, 0, 0);

  // 4) RoPE(q_pe) at position S-1, in place
  {
    int tot = B * HEADS * (ROPE_DIM / 2);
    rope_q<<<(tot + 255) / 256, 256, 0, stream>>>(q_pe, B, S);
  }

  // 5) RoPE(k_pe) for all positions -> kpe (S-1 row from kv_new)
  {
    long tot = (long)B * S * (ROPE_DIM / 2);
    rope_k<<<(int)((tot + 255) / 256), 256, 0, stream>>>(kv_cache, kv_new, kpe, B, S);
  }

  // 6) q_abs[b,h,d] = sum_q q_lora[b,q] * Wqk[h,d,q]  (batched over h)
  gemm_rcr_mt<2><<<dim3(BLKS(WAVES(B, KV_LORA, 2)), HEADS), 128, 0, stream>>>(
      q_lora, Wqk, q_abs, B, KV_LORA, QL,
      QL, QL, HEADS * KV_LORA,
      0, (long)KV_LORA * QL, (long)KV_LORA);

  // 7) scores (fused c_kv + rope parts, scaled); one wave per s-tile, all heads
  mla_scores<<<dim3(BLKS(S / 16), B), 128, 0, stream>>>(
      q_abs, q_pe, kv_cache, kv_new, kpe, scores, S);

  // 8) softmax over s
  softmax_rows<<<B * HEADS, 256, 0, stream>>>(scores, S);

  // 9) o_lat = attn @ c_kv; one wave per d-tile, all heads
  mla_olat<<<dim3(BLKS(KV_LORA / 16), B), 128, 0, stream>>>(
      scores, kv_cache, kv_new, o_lat, S);

  // 10) o[b, h*128+v] = sum_d o_lat[b,h,d] * Wuv[h,d,v]  (batched over h)
  gemm_rrr_mt<2><<<dim3(BLKS(WAVES(B, VDIM, 2)), HEADS), 128, 0, stream>>>(
      o_lat, Wuv, o_cat, B, VDIM, KV_LORA,
      HEADS * KV_LORA, VDIM, HEADS * VDIM,
      (long)KV_LORA, (long)KV_LORA * VDIM, (long)VDIM);

  // 11) out = o @ W_o  (32x16384 @ 16384x7168)
  gemm_rrr_mt<2><<<dim3(BLKS(WAVES(B, DIM, 2)), 1), 128, 0, stream>>>(
      o_cat, Wo, out, B, DIM, HEADS * VDIM,
      HEADS * VDIM, DIM, DIM, 0, 0, 0);
}